// PairwiseAggregator_10771777978672
// MI455X (gfx1250) — compile-verified
//
#include <hip/hip_runtime.h>

// MI455X / gfx1250, wave32. Full-fp32 WMMA pipeline (V_WMMA_F32_16X16X4_F32)
// so numerics match the fp32 reference exactly while still driving the
// matrix pipes.

typedef __attribute__((ext_vector_type(2))) float v2f;
typedef __attribute__((ext_vector_type(8))) float v8f;

#define D_MODEL 24
#define REL_DIM 10
#define NN 512
#define BB 8
// Layer-1 fused K: 24 (|cdiff|) + 24 (x_j) + 4 (qo, cd, ones, pad) = 52 = 13*4
#define K1_CHUNKS 13
// Layer-2 K: 10 padded to 12 = 3*4
#define K2_CHUNKS 3

// One wave per output row (b,i). Tile 16 j's per iteration.
//
// fp32 WMMA operand layouts (CDNA5 ISA 7.12.2):
//   A 16x4:  lane l -> M = l%16, VGPR v holds K = 2*(l/16) + v
//   C/D 16x16: VGPR r on lane l -> element (M = r + 8*(l/16), N = l%16)
//   B 4x16 (assumed mirror of A): lane l, VGPR v -> (K = 2*(l/16)+v, N = l%16)
__global__ __launch_bounds__(32) void pairagg_kernel(
    const float* __restrict__ x, const float* __restrict__ q,
    const float* __restrict__ coord, const float* __restrict__ W1,
    const float* __restrict__ b1, const float* __restrict__ W2,
    const float* __restrict__ b2, float* __restrict__ out)
{
  __shared__ float hT[16 * 16];   // layer-1 output transpose staging

  const int bi   = blockIdx.x;          // 0 .. B*N-1
  const int b    = bi / NN;
  const int i    = bi % NN;
  const int lane = threadIdx.x;         // 0..31
  const int m    = lane & 15;           // A row (j_local) / B & D column
  const int hh   = lane >> 4;           // lane half

  const float* xB = x     + (size_t)b * NN * D_MODEL;
  const float* qB = q     + (size_t)b * NN * D_MODEL;
  const float* cB = coord + (size_t)b * NN * D_MODEL;

  // ---- loop-invariant row-i data, split by the lane-half K pattern ----
  v2f ci[6], qi[6];
#pragma unroll
  for (int c = 0; c < 6; ++c) {
    const int d = 4 * c + 2 * hh;
    ci[c] = *(const v2f*)(cB + i * D_MODEL + d);
    qi[c] = *(const v2f*)(qB + i * D_MODEL + d);
  }

  // Ones-row of B1: b1[n] + x_i @ W1[0:24, n]   (i-only, folded into the GEMM)
  float bias_row = 0.0f;
  if (m < REL_DIM) {
    bias_row = b1[m];
#pragma unroll
    for (int d = 0; d < D_MODEL; ++d)
      bias_row = fmaf(xB[i * D_MODEL + d], W1[d * REL_DIM + m], bias_row);
  }

  // ---- B1 (52 x 16), zero-padded beyond n>=10 ----
  // k<24  -> W1 row 48+k (|cdiff| block) ; 24<=k<48 -> W1 row k (x_j block)
  // k==48 -> W1 row 72 (q_overlap) ; k==49 -> W1 row 73 (coord_dist)
  // k==50 -> ones row (bias_row) ; k==51 -> zero pad
  v2f B1[K1_CHUNKS];
#pragma unroll
  for (int c = 0; c < K1_CHUNKS; ++c) {
#pragma unroll
    for (int v = 0; v < 2; ++v) {
      const int k = 4 * c + 2 * hh + v;
      float val = 0.0f;
      if (m < REL_DIM) {
        int row = -1;
        if      (k < 24)  row = 48 + k;
        else if (k < 48)  row = k;
        else if (k == 48) row = 72;
        else if (k == 49) row = 73;
        if (row >= 0)       val = W1[row * REL_DIM + m];
        else if (k == 50)   val = bias_row;
      }
      B1[c][v] = val;
    }
  }

  // ---- B2 (12 x 16) = W2 zero-padded ----
  v2f B2[K2_CHUNKS];
#pragma unroll
  for (int c = 0; c < K2_CHUNKS; ++c) {
#pragma unroll
    for (int v = 0; v < 2; ++v) {
      const int k = 4 * c + 2 * hh + v;
      B2[c][v] = (k < REL_DIM && m < REL_DIM) ? W2[k * REL_DIM + m] : 0.0f;
    }
  }
  const float b2v = (m < REL_DIM) ? b2[m] : 0.0f;

  float psum = 0.0f;
  float pmax = -1e9f;

  for (int jbase = 0; jbase < NN; jbase += 16) {
    const int j = jbase + m;                 // this lane's A-matrix row
    const float* cj_ptr = cB + j * D_MODEL + 2 * hh;
    const float* xj_ptr = xB + j * D_MODEL + 2 * hh;
    const float* qj_ptr = qB + j * D_MODEL + 2 * hh;

    v2f A[K1_CHUNKS];
    float sqpart = 0.0f, qpart = 0.0f;
#pragma unroll
    for (int c = 0; c < 6; ++c) {
      const v2f cj = *(const v2f*)(cj_ptr + 4 * c);
      const v2f xj = *(const v2f*)(xj_ptr + 4 * c);
      const v2f qj = *(const v2f*)(qj_ptr + 4 * c);
      const float d0 = ci[c][0] - cj[0];
      const float d1 = ci[c][1] - cj[1];
      A[c][0] = fabsf(d0);
      A[c][1] = fabsf(d1);
      sqpart  = fmaf(d0, d0, fmaf(d1, d1, sqpart));
      qpart   = fmaf(qi[c][0], qj[0], fmaf(qi[c][1], qj[1], qpart));
      A[6 + c][0] = xj[0];
      A[6 + c][1] = xj[1];
    }
    // complete the 24-dim reductions across lane halves
    const float qfull = qpart  + __shfl_xor(qpart, 16, 32);
    const float cdist = sqrtf(sqpart + __shfl_xor(sqpart, 16, 32));
    // last K chunk: K=48 -> qo, K=49 -> cd (half 0); K=50 -> 1, K=51 -> 0 (half 1)
    A[12][0] = hh ? 1.0f : qfull;
    A[12][1] = hh ? 0.0f : cdist;

    // ---- layer 1: 13 chained fp32 WMMAs, D = feat @ W1 (+ folded bias) ----
    v8f acc = {};
#pragma unroll
    for (int c = 0; c < K1_CHUNKS; ++c)
      acc = __builtin_amdgcn_wmma_f32_16x16x4_f32(
          false, A[c], false, B1[c], (short)0, acc, false, false);

    // relu + D-layout -> A-layout transpose through LDS
    __syncthreads();
#pragma unroll
    for (int r = 0; r < 8; ++r)
      hT[(r + 8 * hh) * 16 + m] = fmaxf(acc[r], 0.0f);
    __syncthreads();

    v2f A2[K2_CHUNKS];
#pragma unroll
    for (int c = 0; c < K2_CHUNKS; ++c)
      A2[c] = *(const v2f*)(&hT[m * 16 + 4 * c + 2 * hh]);

    // ---- layer 2: 3 chained fp32 WMMAs, rel = h @ W2 ----
    v8f acc2 = {};
#pragma unroll
    for (int c = 0; c < K2_CHUNKS; ++c)
      acc2 = __builtin_amdgcn_wmma_f32_16x16x4_f32(
          false, A2[c], false, B2[c], (short)0, acc2, false, false);

    // relu(+b2), masked accumulate over this tile's 16 rows (lane owns col o=m)
#pragma unroll
    for (int r = 0; r < 8; ++r) {
      const float rel  = fmaxf(acc2[r] + b2v, 0.0f);
      const bool  diag = (jbase + r + 8 * hh) == i;
      psum += diag ? 0.0f : rel;
      pmax  = diag ? pmax : fmaxf(pmax, rel);
    }
  }

  // fold the two lane halves (rows 0..7 vs 8..15 partials)
  psum += __shfl_xor(psum, 16, 32);
  pmax  = fmaxf(pmax, __shfl_xor(pmax, 16, 32));

  if (lane < REL_DIM) {
    const size_t base = ((size_t)b * NN + i) * REL_DIM + lane;
    out[base] = psum / (float)(NN - 1);                       // rel_mean
    out[(size_t)BB * NN * REL_DIM + base] = pmax;             // rel_max
  }
}

extern "C" void kernel_launch(void* const* d_in, const int* in_sizes, int n_in,
                              void* d_out, int out_size, void* d_ws, size_t ws_size,
                              hipStream_t stream) {
  const float* x  = (const float*)d_in[0];
  const float* q  = (const float*)d_in[1];
  const float* co = (const float*)d_in[2];
  const float* W1 = (const float*)d_in[3];
  const float* b1 = (const float*)d_in[4];
  const float* W2 = (const float*)d_in[5];
  const float* b2 = (const float*)d_in[6];
  float* out = (float*)d_out;

  dim3 grid(BB * NN);   // one wave per output row (b, i) -> 4096 waves
  dim3 block(32);
  hipLaunchKernelGGL(pairagg_kernel, grid, block, 0, stream,
                     x, q, co, W1, b1, W2, b2, out);
}